// GCNII_TSC_17609365914389
// MI455X (gfx1250) — compile-verified
//
#include <hip/hip_runtime.h>
#include <cmath>

// ---------------------------------------------------------------------------
// GCNII forward + InfoNCE loss for MI455X (gfx1250), wave32.
// Matrix path: v_wmma_f32_16x16x4_f32 (fp32 in/out, K stepped by 4).
// Data path:   global_load_async_to_lds_b128 double-buffering in the loss GEMM.
// ---------------------------------------------------------------------------

#define N_NODES 8192
#define NFEAT   512
#define NHID    64
#define NCLASS  40
#define NLAYERS 8
#define NEDGE   262144
#define ALPHA   0.1f
#define LAM     0.5f
#define LAMDA   0.5f
// 1/TAU = 2.0

typedef __attribute__((ext_vector_type(2))) float v2f;
typedef __attribute__((ext_vector_type(8))) float v8f;
typedef __attribute__((__vector_size__(16))) int v4i;

// ---- gfx1250 async global->LDS helpers ------------------------------------
#if defined(__has_builtin)
#if __has_builtin(__builtin_amdgcn_global_load_async_to_lds_b128)
#define USE_ASYNC_LDS 1
#endif
#endif
#ifndef USE_ASYNC_LDS
#define USE_ASYNC_LDS 0
#endif

__device__ __forceinline__ void async_copy_b128(const float* g, float* l) {
#if USE_ASYNC_LDS
  __builtin_amdgcn_global_load_async_to_lds_b128(
      (__attribute__((address_space(1))) v4i*)g,
      (__attribute__((address_space(3))) v4i*)l, 0, 0);
#else
  *(float4*)l = *(const float4*)g;
#endif
}

template <int N>
__device__ __forceinline__ void async_wait() {
#if USE_ASYNC_LDS
#if __has_builtin(__builtin_amdgcn_s_wait_asynccnt)
  __builtin_amdgcn_s_wait_asynccnt(N);
#else
  asm volatile("s_wait_asynccnt %0" ::"i"(N) : "memory");
#endif
#endif
}

// ---------------------------------------------------------------------------
// zero fill
__global__ void k_zero(float* __restrict__ p, int n) {
  int i = blockIdx.x * blockDim.x + threadIdx.x;
  if (i < n) p[i] = 0.0f;
}

// ---------------------------------------------------------------------------
// h0 = relu(x @ W0 + b0)   [8192,512]x[512,64]
// grid(512,4), block(32): one wave -> one 16x16 output tile, K accumulated 4 at a time.
__global__ void k_gemm_h0(const float* __restrict__ x, const float* __restrict__ W0,
                          const float* __restrict__ b0, float* __restrict__ h0) {
  const int m0   = blockIdx.x * 16;
  const int n0   = blockIdx.y * 16;
  const int lane = threadIdx.x;
  const int half = lane >> 4;   // 0: K+0/K+1, 1: K+2/K+3
  const int idx  = lane & 15;   // row (A) / col (B)

  v8f c = {};
  for (int k0 = 0; k0 < NFEAT; k0 += 4) {
    if ((k0 & 63) == 0 && k0 + 64 < NFEAT)
      __builtin_prefetch(&x[(m0 + idx) * NFEAT + k0 + 64], 0, 1);
    const int kk = k0 + 2 * half;
    v2f a, b;
    a.x = x[(m0 + idx) * NFEAT + kk];
    a.y = x[(m0 + idx) * NFEAT + kk + 1];
    b.x = W0[(kk    ) * NHID + n0 + idx];
    b.y = W0[(kk + 1) * NHID + n0 + idx];
    c = __builtin_amdgcn_wmma_f32_16x16x4_f32(false, a, false, b, (short)0, c,
                                              false, false);
  }
  const int col  = n0 + idx;
  const float bv = b0[col];
#pragma unroll
  for (int r = 0; r < 8; ++r) {
    const int row = m0 + r + 8 * half;
    float v = c[r] + bv;
    h0[row * NHID + col] = v > 0.0f ? v : 0.0f;
  }
}

// ---------------------------------------------------------------------------
// hi[dst] += w * h[src]  (scatter-add SPMM). One thread per (edge, feature).
__global__ void k_spmm(const int* __restrict__ src, const int* __restrict__ dst,
                       const float* __restrict__ w, const float* __restrict__ h,
                       float* __restrict__ hi) {
  const int t = blockIdx.x * blockDim.x + threadIdx.x;
  const int e = t >> 6;
  const int f = t & 63;
  if (e < NEDGE) {
    const float v = w[e] * h[src[e] * NHID + f];
    atomicAdd(&hi[dst[e] * NHID + f], v);
  }
}

// ---------------------------------------------------------------------------
// One GCNII layer, fused:
//   S   = (1-a)*hi + a*h0
//   cur = relu(theta*(S@Wc) + (1-theta)*S)
//   h   = (1-beta)*cur + beta*h
// grid(512), block(32): one wave -> 16 rows x full 64 cols (4 accumulators).
__global__ void k_layer(const float* __restrict__ hi, const float* __restrict__ h0,
                        float* __restrict__ h, const float* __restrict__ Wc,
                        float theta, float beta) {
  __shared__ float S[16][NHID];
  const int m0   = blockIdx.x * 16;
  const int lane = threadIdx.x;

  for (int i = lane; i < 16 * NHID; i += 32) {
    const int r = i >> 6, cidx = i & 63;
    S[r][cidx] = (1.0f - ALPHA) * hi[(m0 + r) * NHID + cidx]
               + ALPHA * h0[(m0 + r) * NHID + cidx];
  }
  __syncthreads();

  const int half = lane >> 4;
  const int idx  = lane & 15;

  v8f acc[4] = {v8f{}, v8f{}, v8f{}, v8f{}};
  for (int k0 = 0; k0 < NHID; k0 += 4) {
    const int kk = k0 + 2 * half;
    v2f a;
    a.x = S[idx][kk];
    a.y = S[idx][kk + 1];
#pragma unroll
    for (int nt = 0; nt < 4; ++nt) {
      v2f b;
      b.x = Wc[(kk    ) * NHID + nt * 16 + idx];
      b.y = Wc[(kk + 1) * NHID + nt * 16 + idx];
      acc[nt] = __builtin_amdgcn_wmma_f32_16x16x4_f32(false, a, false, b,
                                                      (short)0, acc[nt],
                                                      false, false);
    }
  }

#pragma unroll
  for (int nt = 0; nt < 4; ++nt) {
    const int col = nt * 16 + idx;
#pragma unroll
    for (int r = 0; r < 8; ++r) {
      const int row   = m0 + r + 8 * half;
      const float sup = S[r + 8 * half][col];
      float cur = theta * acc[nt][r] + (1.0f - theta) * sup;
      cur = cur > 0.0f ? cur : 0.0f;
      h[row * NHID + col] = (1.0f - beta) * cur + beta * h[row * NHID + col];
    }
  }
}

// ---------------------------------------------------------------------------
// z[i] = h[i] / ||h[i]||    grid(8192), block(32)
__global__ void k_norm(const float* __restrict__ h, float* __restrict__ z) {
  const int row  = blockIdx.x;
  const int lane = threadIdx.x;
  const float a = h[row * NHID + lane];
  const float b = h[row * NHID + 32 + lane];
  float ss = a * a + b * b;
#pragma unroll
  for (int m = 16; m >= 1; m >>= 1) ss += __shfl_xor(ss, m, 32);
  const float inv = rsqrtf(ss);
  z[row * NHID + lane]      = a * inv;
  z[row * NHID + 32 + lane] = b * inv;
}

// ---------------------------------------------------------------------------
// InfoNCE loss (z1 == z2):
//   S = z z^T (tiled, never materialized), rowsum_i = sum_j exp(2*S_ij),
//   ct_i = log(2*(rowsum_i - diag_i)) - log(diag_i), loss = mean(ct)
// grid(512), block(256) = 8 waves; 16 target rows cached in LDS; column data
// streamed through a double-buffered LDS chunk (128 cols = 32KB) filled with
// global_load_async_to_lds_b128 and synchronized via s_wait_asynccnt.
#define CHUNK   128
#define NCHUNK  (N_NODES / CHUNK)   // 64
#define CPY_PER_THREAD ((CHUNK * NHID) / (256 * 4))  // 8 x b128 per thread

__global__ void k_loss(const float* __restrict__ z, float* __restrict__ loss) {
  __shared__ float Arows[16][NHID];
  __shared__ float colbuf[2][CHUNK * NHID];
  __shared__ float srow[16];
  __shared__ float sdiag[16];

  const int m0   = blockIdx.x * 16;
  const int tid  = threadIdx.x;
  const int wave = tid >> 5;
  const int lane = tid & 31;

  for (int i = tid; i < 16 * NHID; i += 256) {
    Arows[i >> 6][i & 63] = z[(m0 + (i >> 6)) * NHID + (i & 63)];
  }
  if (tid < 16) { srow[tid] = 0.0f; sdiag[tid] = 0.0f; }

  // kick off chunk 0 -> buffer 0
#pragma unroll
  for (int i = 0; i < CPY_PER_THREAD; ++i) {
    const int off = (tid + i * 256) * 4;
    async_copy_b128(z + off, &colbuf[0][off]);
  }

  const int half = lane >> 4;
  const int idx  = lane & 15;

  float rs[8] = {0, 0, 0, 0, 0, 0, 0, 0};
  float dg[8] = {0, 0, 0, 0, 0, 0, 0, 0};

  for (int c = 0; c < NCHUNK; ++c) {
    if (c + 1 < NCHUNK) {
      // prefetch next chunk into the other buffer while this one computes
      const float* g = z + (size_t)(c + 1) * CHUNK * NHID;
      float* dstb = colbuf[(c + 1) & 1];
#pragma unroll
      for (int i = 0; i < CPY_PER_THREAD; ++i) {
        const int off = (tid + i * 256) * 4;
        async_copy_b128(g + off, dstb + off);
      }
      async_wait<CPY_PER_THREAD>();   // chunk c complete (in-order), c+1 in flight
    } else {
      async_wait<0>();
    }
    __syncthreads();                  // chunk c visible to all waves

    const float* B = colbuf[c & 1] + (wave * 16) * NHID;  // this wave's 16 cols
    v8f acc = {};
    for (int k0 = 0; k0 < NHID; k0 += 4) {
      const int kk = k0 + 2 * half;
      v2f a, b;
      a.x = Arows[idx][kk];
      a.y = Arows[idx][kk + 1];
      b.x = B[idx * NHID + kk];
      b.y = B[idx * NHID + kk + 1];
      acc = __builtin_amdgcn_wmma_f32_16x16x4_f32(false, a, false, b, (short)0,
                                                  acc, false, false);
    }

    const int col = c * CHUNK + wave * 16 + idx;
#pragma unroll
    for (int r = 0; r < 8; ++r) {
      const float e = expf(acc[r] * 2.0f);   // 1/TAU = 2
      rs[r] += e;
      if ((m0 + r + 8 * half) == col) dg[r] += e;
    }
    __syncthreads();                  // done reading before buffer reuse
  }

  // reduce over the 16 columns held by each lane-half (masks <= 8 stay in-half)
#pragma unroll
  for (int r = 0; r < 8; ++r) {
    float v = rs[r], d = dg[r];
#pragma unroll
    for (int m = 8; m >= 1; m >>= 1) {
      v += __shfl_xor(v, m, 32);
      d += __shfl_xor(d, m, 32);
    }
    if (idx == 0) {
      atomicAdd(&srow[r + 8 * half], v);
      atomicAdd(&sdiag[r + 8 * half], d);
    }
  }
  __syncthreads();

  if (tid < 16) {
    const float diag  = sdiag[tid];
    const float denom = 2.0f * (srow[tid] - diag);
    const float ct    = logf(denom) - logf(diag);
    atomicAdd(loss, ct * (1.0f / (float)N_NODES));
  }
}

// ---------------------------------------------------------------------------
// out = log_softmax(h @ W1 + b1)   grid(8192), block(64)
__global__ void k_logits(const float* __restrict__ h, const float* __restrict__ W1,
                         const float* __restrict__ b1, float* __restrict__ out) {
  __shared__ float hrow[NHID];
  __shared__ float lg[NCLASS];
  __shared__ float s_lse;

  const int row = blockIdx.x;
  const int t   = threadIdx.x;
  hrow[t] = h[row * NHID + t];
  __syncthreads();

  if (t < NCLASS) {
    float v = b1[t];
    for (int k = 0; k < NHID; ++k) v += hrow[k] * W1[k * NCLASS + t];
    lg[t] = v;
  }
  __syncthreads();

  if (t == 0) {
    float mx = lg[0];
    for (int i = 1; i < NCLASS; ++i) mx = lg[i] > mx ? lg[i] : mx;
    float s = 0.0f;
    for (int i = 0; i < NCLASS; ++i) s += expf(lg[i] - mx);
    s_lse = mx + logf(s);
  }
  __syncthreads();

  if (t < NCLASS) out[row * NCLASS + t] = lg[t] - s_lse;
}

// ---------------------------------------------------------------------------
extern "C" void kernel_launch(void* const* d_in, const int* in_sizes, int n_in,
                              void* d_out, int out_size, void* d_ws, size_t ws_size,
                              hipStream_t stream) {
  (void)in_sizes; (void)n_in; (void)out_size; (void)ws_size;

  const float* x     = (const float*)d_in[0];
  const float* W0    = (const float*)d_in[1];
  const float* b0    = (const float*)d_in[2];
  const float* convw = (const float*)d_in[3];
  const float* W1    = (const float*)d_in[4];
  const float* b1    = (const float*)d_in[5];
  const float* ew    = (const float*)d_in[6];
  const int*   ei    = (const int*)d_in[7];
  const int*   src   = ei;           // edge_index[0]
  const int*   dst   = ei + NEDGE;   // edge_index[1]

  const int NNH = N_NODES * NHID;    // 524288 floats per buffer
  float* h0 = (float*)d_ws;
  float* h  = h0 + NNH;
  float* hi = h  + NNH;              // reused as z after the layer loop

  float* out  = (float*)d_out;
  float* loss = out + N_NODES * NCLASS;

  k_zero<<<1, 64, 0, stream>>>(loss, 1);

  // h0 = relu(x @ W0 + b0); h = h0
  k_gemm_h0<<<dim3(N_NODES / 16, NHID / 16), 32, 0, stream>>>(x, W0, b0, h0);
  (void)hipMemcpyAsync(h, h0, (size_t)NNH * sizeof(float),
                       hipMemcpyDeviceToDevice, stream);

  for (int i = 0; i < NLAYERS; ++i) {
    const int l = i + 1;
    const float theta = logf(LAM / (float)l + 1.0f);
    const float beta  = LAMDA / (float)l;
    k_zero<<<NNH / 256, 256, 0, stream>>>(hi, NNH);
    k_spmm<<<(NEDGE * 64) / 256, 256, 0, stream>>>(src, dst, ew, h, hi);
    k_layer<<<N_NODES / 16, 32, 0, stream>>>(hi, h0, h, convw + i * NHID * NHID,
                                             theta, beta);
  }

  float* zbuf = hi;  // hi no longer needed
  k_norm<<<N_NODES, 32, 0, stream>>>(h, zbuf);
  k_loss<<<N_NODES / 16, 256, 0, stream>>>(zbuf, loss);
  k_logits<<<N_NODES, 64, 0, stream>>>(h, W1, b1, out);
}